// VoxelHashTable_2499670966393
// MI455X (gfx1250) — compile-verified
//
#include <hip/hip_runtime.h>
#include <cstdint>
#include <cstddef>

// (PRIMES % TSIZE) as uint32 — TSIZE = 2^20, so all mod-2^20 math can be done
// in wrapping uint32 arithmetic followed by & 0xFFFFF (2^20 divides 2^32).
#define PMX 455773u
#define PMY 475301u
#define PMZ 655287u
#define HMASK 0xFFFFFu
#define FDIM 32
#define BLK 128
#define ROWPAD 36   // 32 floats + 4 pad: breaks 128B-stride LDS bank conflicts, keeps 16B align

#if __has_builtin(__builtin_amdgcn_global_load_async_to_lds_b128) && \
    __has_builtin(__builtin_amdgcn_s_wait_asynccnt)
#define USE_ASYNC 1
#else
#define USE_ASYNC 0
#endif

typedef float f4  __attribute__((ext_vector_type(4)));
typedef int   v4i __attribute__((ext_vector_type(4)));

#if USE_ASYNC
// Builtin signature (from hipcc diagnostic): param 1 is AS1 pointer-to-int4
// (clang prints AS1 as '__device__'); param 2 is the LDS-side int4 pointer.
typedef __attribute__((address_space(1))) v4i as1_v4i;
typedef __attribute__((address_space(3))) v4i as3_v4i;

// One feature row = 128 B = 8 x b128 async transfers. Offset baked into both
// pointers (hardware applies IOFFSET to both sides; we just pass 0).
__device__ __forceinline__ void issue_row_async(const float* __restrict__ g, float* l) {
#pragma unroll
    for (int o = 0; o < 8; ++o) {
        __builtin_amdgcn_global_load_async_to_lds_b128(
            (as1_v4i*)(void*)((const char*)g + 16 * o),
            (as3_v4i*)(void*)((char*)l + 16 * o),
            0, 0);
    }
}
#endif

__global__ __launch_bounds__(BLK) void voxel_hash_query(
    const float* __restrict__ qp,
    const float* __restrict__ feats0,
    const float* __restrict__ feats1,
    const int*   __restrict__ h2v0,
    const int*   __restrict__ h2v1,
    float* __restrict__ out, int M)
{
#if USE_ASYNC
    __shared__ float stage[2][BLK][ROWPAD];
#endif
    const int tid = threadIdx.x;
    const int t   = blockIdx.x * BLK + tid;
    if (M <= 0) return;
    const int qi  = (t < M) ? t : (M - 1);   // clamp tail; stores are guarded

    const float qx = qp[3 * qi + 0];
    const float qy = qp[3 * qi + 1];
    const float qz = qp[3 * qi + 2];

    float        w[16];
    const float* rp[16];

    // ---- setup: both levels' hash lookups + trilinear weights (16 h2v loads in flight) ----
#pragma unroll
    for (int l = 0; l < 2; ++l) {
        const float res = l ? 0.24f : 0.12f;
        const float* __restrict__ F = l ? feats1 : feats0;
        const int*   __restrict__ H = l ? h2v1  : h2v0;
        const float fx = qx / res, fy = qy / res, fz = qz / res;
        const float bx = floorf(fx), by = floorf(fy), bz = floorf(fz);
        const float tx = fx - bx, ty = fy - by, tz = fz - bz;
        const uint32_t s = (uint32_t)(int)bx * PMX +
                           (uint32_t)(int)by * PMY +
                           (uint32_t)(int)bz * PMZ;
#pragma unroll
        for (int c = 0; c < 8; ++c) {
            const uint32_t h = (s + ((c & 4) ? PMX : 0u)
                                  + ((c & 2) ? PMY : 0u)
                                  + ((c & 1) ? PMZ : 0u)) & HMASK;
            const int v = H[h];
            const float wc = ((c & 4) ? tx : 1.0f - tx) *
                             ((c & 2) ? ty : 1.0f - ty) *
                             ((c & 1) ? tz : 1.0f - tz);
            const int k = l * 8 + c;
            w[k]  = (v >= 0) ? wc : 0.0f;                       // invalid corner -> 0 weight
            rp[k] = F + (size_t)((v >= 0) ? v : 0) * FDIM;      // invalid -> row 0 (harmless)
        }
    }

    // warm the rows (lowers to global_prefetch_b8 on gfx1250)
#pragma unroll
    for (int k = 0; k < 16; ++k) __builtin_prefetch(rp[k]);

    f4 acc[8];
#pragma unroll
    for (int j = 0; j < 8; ++j) acc[j] = (f4)0.0f;

    auto store_half = [&](int half) {
        if (t < M) {
            f4* ob = (f4*)(out + (size_t)t * 64 + half * 32);
#pragma unroll
            for (int j = 0; j < 8; ++j) ob[j] = acc[j];
        }
    };

#if USE_ASYNC
    // ---- double-buffered async gather pipeline over 16 corner rows ----
    issue_row_async(rp[0], &stage[0][tid][0]);
#pragma unroll
    for (int c = 0; c < 16; ++c) {
        if (c + 1 < 16) {
            issue_row_async(rp[c + 1], &stage[(c + 1) & 1][tid][0]);
            __builtin_amdgcn_s_wait_asynccnt(8);   // corner c's 8 xfers done (in-order)
        } else {
            __builtin_amdgcn_s_wait_asynccnt(0);
        }
        __asm__ volatile("" ::: "memory");          // don't hoist LDS reads above the wait
        const f4* r4 = (const f4*)&stage[c & 1][tid][0];
        const float wc = w[c];
#pragma unroll
        for (int j = 0; j < 8; ++j) acc[j] += wc * r4[j];
        if (c == 7) {
            store_half(0);
#pragma unroll
            for (int j = 0; j < 8; ++j) acc[j] = (f4)0.0f;
        }
    }
    store_half(1);
#else
    // ---- fallback: direct wide gathers ----
#pragma unroll
    for (int c = 0; c < 16; ++c) {
        const f4* r4 = (const f4*)rp[c];
        const float wc = w[c];
#pragma unroll
        for (int j = 0; j < 8; ++j) acc[j] += wc * r4[j];
        if (c == 7) {
            store_half(0);
#pragma unroll
            for (int j = 0; j < 8; ++j) acc[j] = (f4)0.0f;
        }
    }
    store_half(1);
#endif
}

extern "C" void kernel_launch(void* const* d_in, const int* in_sizes, int n_in,
                              void* d_out, int out_size, void* d_ws, size_t ws_size,
                              hipStream_t stream) {
    const float* qp = (const float*)d_in[0];
    const float* f0 = (const float*)d_in[1];
    const float* f1 = (const float*)d_in[2];
    const int*   h0 = (const int*)d_in[3];
    const int*   h1 = (const int*)d_in[4];
    float* out = (float*)d_out;
    const int M = in_sizes[0] / 3;
    const int grid = (M + BLK - 1) / BLK;
    hipLaunchKernelGGL(voxel_hash_query, dim3(grid), dim3(BLK), 0, stream,
                       qp, f0, f1, h0, h1, out, M);
}